// CNAPSProtoNetSimilarityModule_53541062311980
// MI455X (gfx1250) — compile-verified
//
#include <hip/hip_runtime.h>
#include <math.h>

// Problem constants (from reference setup_inputs)
#define B_ 256
#define S_ 512
#define Q_ 256
#define D_ 512

typedef __attribute__((ext_vector_type(2))) float v2f;
typedef __attribute__((ext_vector_type(8))) float v8f;

// ---------------------------------------------------------------------------
// Kernel 1: per-batch masks, positive count, and class means.
// One block per batch, 512 threads (thread == dimension d == support index s).
// ---------------------------------------------------------------------------
__global__ __launch_bounds__(512)
void k_stats(const float* __restrict__ sup, const int* __restrict__ labels,
             const int* __restrict__ slen,
             float* __restrict__ pmask, float* __restrict__ vmask,
             float* __restrict__ mu_pos, float* __restrict__ mu_neg,
             float* __restrict__ mu_all, float* __restrict__ npos_arr)
{
    int b   = blockIdx.x;
    int tid = threadIdx.x;
    __shared__ float pm[S_];
    __shared__ float vm[S_];
    __shared__ float red[512];

    int len = slen[b];
    {
        int s   = tid;
        int lab = labels[b * S_ + s];
        float v = (s < len) ? 1.f : 0.f;
        float p = (lab == 1) ? v : 0.f;
        pm[s] = p; vm[s] = v;
        pmask[b * S_ + s] = p;
        vmask[b * S_ + s] = v;
        red[tid] = p;
    }
    __syncthreads();
    for (int off = 256; off > 0; off >>= 1) {
        if (tid < off) red[tid] += red[tid + off];
        __syncthreads();
    }
    float np = red[0];
    float n  = (float)len;
    float nn = n - np;
    if (tid == 0) npos_arr[b] = np;

    // column sums: thread tid == dimension d (coalesced across threads)
    int d = tid;
    float sp = 0.f, sa = 0.f;
    const float* xb = sup + (size_t)b * S_ * D_ + d;
    for (int s = 0; s < S_; ++s) {
        float x = xb[(size_t)s * D_];
        sp = fmaf(x, pm[s], sp);
        sa = fmaf(x, vm[s], sa);
    }
    mu_pos[b * D_ + d] = sp / np;
    mu_all[b * D_ + d] = sa / n;
    mu_neg[b * D_ + d] = (sa - sp) / nn;
}

// ---------------------------------------------------------------------------
// Kernel 2: Gram matrices via V_WMMA_F32_16X16X4_F32 and direct assembly of
//   A_pos = 0.1*pos_cov + 0.9*task_cov + 0.1*I
//   A_neg = 0.1*neg_cov + 0.9*task_cov + 0.1*I
// using (n-1)*cov = G - n*mu*mu^T and G_neg = G_all - G_pos.
// One wave (32 threads) per upper-triangular 16x16 tile; mirror writes.
// A operand is mask-scaled on load (X~^T X~ == X^T M X since mask^2 == mask).
// ---------------------------------------------------------------------------
__global__ __launch_bounds__(32)
void k_gram(const float* __restrict__ sup,
            const float* __restrict__ pmask, const float* __restrict__ vmask,
            const float* __restrict__ mu_pos, const float* __restrict__ mu_neg,
            const float* __restrict__ mu_all,
            const float* __restrict__ npos_arr, const int* __restrict__ slen,
            float* __restrict__ Apos, float* __restrict__ Aneg)
{
    int b = blockIdx.y;
    // decode upper-triangle tile index -> (ti, tj), tj >= ti, 32x32 tiles
    int t = blockIdx.x;
    int ti = 0;
    while (t >= (32 - ti)) { t -= (32 - ti); ++ti; }
    int tj = ti + t;
    int i0 = ti * 16, j0 = tj * 16;

    int lane = threadIdx.x;
    int l = lane & 15, h = lane >> 4;

    const float* Xb  = sup   + (size_t)b * S_ * D_;
    const float* pmb = pmask + b * S_;
    const float* vmb = vmask + b * S_;

    v8f accP = {}; // positive-masked Gram
    v8f accA = {}; // valid-masked Gram
    for (int k0 = 0; k0 < S_; k0 += 4) {
        int s0 = k0 + 2 * h;   // k = 2h + vgpr  (ISA 16x4 fp32 layout)
        int s1 = s0 + 1;
        float a0 = Xb[(size_t)s0 * D_ + i0 + l];
        float a1 = Xb[(size_t)s1 * D_ + i0 + l];
        v2f bv; bv.x = Xb[(size_t)s0 * D_ + j0 + l];
                bv.y = Xb[(size_t)s1 * D_ + j0 + l];
        float p0 = pmb[s0], p1 = pmb[s1];
        float v0 = vmb[s0], v1 = vmb[s1];
        v2f ap; ap.x = a0 * p0; ap.y = a1 * p1;
        v2f av; av.x = a0 * v0; av.y = a1 * v1;
        accP = __builtin_amdgcn_wmma_f32_16x16x4_f32(false, ap, false, bv,
                                                     (short)0, accP, false, false);
        accA = __builtin_amdgcn_wmma_f32_16x16x4_f32(false, av, false, bv,
                                                     (short)0, accA, false, false);
    }

    float np = npos_arr[b];
    float n  = (float)slen[b];
    float nn = n - np;
    int jj = j0 + l;
    float mupJ = mu_pos[b * D_ + jj];
    float munJ = mu_neg[b * D_ + jj];
    float muaJ = mu_all[b * D_ + jj];
    float* Ap = Apos + (size_t)b * D_ * D_;
    float* An = Aneg + (size_t)b * D_ * D_;

    for (int r = 0; r < 8; ++r) {
        int i = i0 + r + 8 * h;            // C/D layout: M = r + 8h, N = l
        float mupI = mu_pos[b * D_ + i];
        float munI = mu_neg[b * D_ + i];
        float muaI = mu_all[b * D_ + i];
        float gp = accP[r];
        float ga = accA[r];
        float gn = ga - gp;
        float covP = (gp - np * mupI * mupJ) / (np - 1.f);
        float covN = (gn - nn * munI * munJ) / (nn - 1.f);
        float covT = (ga - n  * muaI * muaJ) / (n  - 1.f);
        float diag = (i == jj) ? 0.1f : 0.f;
        float apv = 0.1f * covP + 0.9f * covT + diag;
        float anv = 0.1f * covN + 0.9f * covT + diag;
        Ap[(size_t)i * D_ + jj] = apv;
        An[(size_t)i * D_ + jj] = anv;
        if (ti != tj) {                    // symmetric mirror
            Ap[(size_t)jj * D_ + i] = apv;
            An[(size_t)jj * D_ + i] = anv;
        }
    }
}

// ---------------------------------------------------------------------------
// Kernel 3: BLOCKED in-place Gauss-Jordan inversion (SPD + ridge, no pivoting).
// One 512-thread block per 512x512 matrix, 32 panel steps of width 16.
// Per step: stage panel A[:,k0:k0+16] (stride-17 LDS) and the ORIGINAL 16
// pivot rows R (stride-520 LDS); run 16 GJ pivot steps inside the panel
// (self-contained; the panel becomes the block columns of the combined
// elimination matrix M); then apply M to all other columns as a rank-16
// trailing GEMM on WMMA:  A[i,j] <- (i notin block ? A[i,j] : 0)
//                                   + sum_b panel[i,b] * R[b,j]
// and finally write the panel back as the new block columns.
// Traffic: 32 sweeps/matrix instead of 512; 137 GFLOP moved onto WMMA.
// ---------------------------------------------------------------------------
#define PS 17    // panel row stride (floats): conflict-free A-operand reads
#define RS 520   // R row stride (floats): separates half-wave bank sets

__global__ __launch_bounds__(512)
void k_invert_blk(float* __restrict__ mats)
{
    float* A = mats + (size_t)blockIdx.x * D_ * D_;
    int tid = threadIdx.x;

    __shared__ float panel[D_ * PS];   // 512 x 16 panel, padded  (34.8 KB)
    __shared__ float Rrow[16 * RS];    // 16 x 512 pivot rows, padded (33.3 KB)

    int lane = tid & 31;
    int wv   = tid >> 5;               // wave id 0..15
    int l = lane & 15, h = lane >> 4;

    for (int kb = 0; kb < D_ / 16; ++kb) {
        int k0 = kb * 16;

        // ---- stage original pivot rows and panel columns ----
        for (int idx = tid; idx < 16 * D_; idx += 512) {
            int bb = idx >> 9, j = idx & (D_ - 1);
            Rrow[bb * RS + j] = A[(size_t)(k0 + bb) * D_ + j];
        }
        for (int idx = tid; idx < D_ * 16; idx += 512) {
            int i = idx >> 4, c = idx & 15;
            panel[i * PS + c] = A[(size_t)i * D_ + k0 + c];
        }
        __syncthreads();

        // ---- 16 in-panel Gauss-Jordan pivot steps (thread == row) ----
        for (int kk = 0; kk < 16; ++kk) {
            int pr = k0 + kk;
            float prow[16];
            #pragma unroll
            for (int c = 0; c < 16; ++c) prow[c] = panel[pr * PS + c];
            float p = 1.0f / prow[kk];
            float f = panel[tid * PS + kk];
            __syncthreads();               // all reads of pivot row done
            if (tid == pr) {
                #pragma unroll
                for (int c = 0; c < 16; ++c)
                    panel[tid * PS + c] = (c == kk) ? p : prow[c] * p;
            } else {
                #pragma unroll
                for (int c = 0; c < 16; ++c) {
                    float sc  = (c == kk) ? p   : prow[c] * p;
                    float own = (c == kk) ? 0.f : panel[tid * PS + c];
                    panel[tid * PS + c] = own - f * sc;
                }
            }
            __syncthreads();
        }

        // ---- trailing rank-16 update via WMMA: 16x16 tiles, K=16 ----
        for (int tt = wv; tt < (D_ / 16) * (D_ / 16); tt += 16) {
            int ti_ = tt >> 5, tj_ = tt & 31;
            if (tj_ == kb) continue;           // block columns replaced below
            int i0 = ti_ * 16, j0 = tj_ * 16;
            v8f acc = {};
            if (ti_ != kb) {                   // identity contribution
                #pragma unroll
                for (int r = 0; r < 8; ++r)
                    acc[r] = A[(size_t)(i0 + r + 8 * h) * D_ + j0 + l];
            }
            #pragma unroll
            for (int kc = 0; kc < 4; ++kc) {
                int c0 = 4 * kc + 2 * h;       // k = 2h + vgpr within chunk
                v2f a;  a.x  = panel[(i0 + l) * PS + c0];
                        a.y  = panel[(i0 + l) * PS + c0 + 1];
                v2f bv; bv.x = Rrow[c0 * RS + j0 + l];
                        bv.y = Rrow[(c0 + 1) * RS + j0 + l];
                acc = __builtin_amdgcn_wmma_f32_16x16x4_f32(false, a, false, bv,
                                                            (short)0, acc,
                                                            false, false);
            }
            #pragma unroll
            for (int r = 0; r < 8; ++r)
                A[(size_t)(i0 + r + 8 * h) * D_ + j0 + l] = acc[r];
        }
        __syncthreads();

        // ---- write panel back as the new block columns ----
        for (int idx = tid; idx < D_ * 16; idx += 512) {
            int i = idx >> 4, c = idx & 15;
            A[(size_t)i * D_ + k0 + c] = panel[i * PS + c];
        }
        __syncthreads();
    }
}

// ---------------------------------------------------------------------------
// Kernel 4: Mahalanobis logits. One wave per (batch, class, 16-query tile).
// diff = mu - query staged in padded LDS (row stride 516 -> conflict-free);
// T = diff @ Inv via WMMA fp32; fused elementwise-mul + rowsum; half-wave
// shuffle reduction; epilogue writes -scale^2 * maha * q_valid to [B,Q,2].
// ---------------------------------------------------------------------------
__global__ __launch_bounds__(32)
void k_maha(const float* __restrict__ query, const float* __restrict__ mu_pos,
            const float* __restrict__ mu_neg, const float* __restrict__ inv_mats,
            const int* __restrict__ qlen, const float* __restrict__ logscale,
            float* __restrict__ out)
{
    int b   = blockIdx.z;
    int cls = blockIdx.y;                 // 0 = neg, 1 = pos (output stack order)
    int q0  = blockIdx.x * 16;
    int lane = threadIdx.x;
    const int LS = D_ + 4;                // 516: pad to avoid LDS bank conflicts
    __shared__ float diff[16 * (D_ + 4)];

    const float* mu  = (cls ? mu_pos : mu_neg) + b * D_;
    const float* Inv = inv_mats + ((size_t)(cls ? b : (B_ + b))) * D_ * D_;
    const float* qb  = query + ((size_t)b * Q_ + q0) * D_;

    for (int q = 0; q < 16; ++q)
        for (int d = lane; d < D_; d += 32)
            diff[q * LS + d] = mu[d] - qb[(size_t)q * D_ + d];
    __syncthreads();

    int l = lane & 15, h = lane >> 4;
    float partial[8] = {0.f, 0.f, 0.f, 0.f, 0.f, 0.f, 0.f, 0.f};

    for (int n0 = 0; n0 < D_; n0 += 16) {
        v8f acc = {};
        for (int k0 = 0; k0 < D_; k0 += 4) {
            int s0 = k0 + 2 * h;
            v2f a;  a.x  = diff[l * LS + s0];
                    a.y  = diff[l * LS + s0 + 1];
            v2f bv; bv.x = Inv[(size_t)s0 * D_ + n0 + l];
                    bv.y = Inv[(size_t)(s0 + 1) * D_ + n0 + l];
            acc = __builtin_amdgcn_wmma_f32_16x16x4_f32(false, a, false, bv,
                                                        (short)0, acc, false, false);
        }
        #pragma unroll
        for (int r = 0; r < 8; ++r) {
            int m = r + 8 * h;                       // local query index
            partial[r] = fmaf(acc[r], diff[m * LS + n0 + l], partial[r]);
        }
    }

    float sc = expf(logscale[0]);
    float neg_s2 = -sc * sc;
    int len = qlen[b];
    #pragma unroll
    for (int r = 0; r < 8; ++r) {
        float v = partial[r];
        v += __shfl_xor(v, 1, 32);   // reduce within each 16-lane half
        v += __shfl_xor(v, 2, 32);
        v += __shfl_xor(v, 4, 32);
        v += __shfl_xor(v, 8, 32);
        if (l == 0) {
            int q = q0 + r + 8 * h;
            float valid = (q < len) ? 1.f : 0.f;
            out[((size_t)b * Q_ + q) * 2 + cls] = neg_s2 * v * valid;
        }
    }
}

// ---------------------------------------------------------------------------
// Host launcher
// ---------------------------------------------------------------------------
extern "C" void kernel_launch(void* const* d_in, const int* in_sizes, int n_in,
                              void* d_out, int out_size, void* d_ws, size_t ws_size,
                              hipStream_t stream)
{
    const float* sup      = (const float*)d_in[0];
    const int*   labels   = (const int*)d_in[1];
    const float* query    = (const float*)d_in[2];
    const int*   slen     = (const int*)d_in[3];
    const int*   qlen     = (const int*)d_in[4];
    const float* logscale = (const float*)d_in[5];
    float* out = (float*)d_out;

    // workspace layout (floats): A_pos[B,D,D] | A_neg[B,D,D] | mu_pos | mu_neg
    //                            | mu_all | npos[B] | pmask[B,S] | vmask[B,S]
    float* ws = (float*)d_ws;
    const size_t BDD = (size_t)B_ * D_ * D_;
    float* Apos   = ws;
    float* Aneg   = ws + BDD;
    float* mu_pos = ws + 2 * BDD;
    float* mu_neg = mu_pos + (size_t)B_ * D_;
    float* mu_all = mu_neg + (size_t)B_ * D_;
    float* npos   = mu_all + (size_t)B_ * D_;
    float* pmask  = npos + B_;
    float* vmask  = pmask + (size_t)B_ * S_;

    k_stats<<<B_, 512, 0, stream>>>(sup, labels, slen, pmask, vmask,
                                    mu_pos, mu_neg, mu_all, npos);

    dim3 g2(528, B_);   // 32*33/2 upper-triangular tiles per batch
    k_gram<<<g2, 32, 0, stream>>>(sup, pmask, vmask, mu_pos, mu_neg, mu_all,
                                  npos, slen, Apos, Aneg);

    k_invert_blk<<<2 * B_, 512, 0, stream>>>(Apos);   // Apos/Aneg contiguous

    dim3 g4(Q_ / 16, 2, B_);
    k_maha<<<g4, 32, 0, stream>>>(query, mu_pos, mu_neg, Apos, qlen, logscale, out);
}